// SelfEmbedding_75977971466555
// MI455X (gfx1250) — compile-verified
//
#include <hip/hip_runtime.h>

// ---------------------------------------------------------------------------
// SelfEmbedding on gfx1250 (CDNA5, wave32, WMMA f32 16x16x4)
//
// Factorized form: cat(xi, self) @ W == xi @ W[:L] + self @ W[L:], and the
// self-term is shared across all K sub-rows of a group -> compute once per
// batch-row tile as a WMMA accumulator (bias folded in) and reuse it as the
// C operand of the per-k small GEMMs.
//
// Streaming traffic (x in: 92 MB, out: 528 MB) is tagged non-temporal so the
// only resident working set in L2/WGP$ is the ~123 KB of weights that every
// workgroup re-reads.
// ---------------------------------------------------------------------------

typedef __attribute__((ext_vector_type(2))) float v2f;
typedef __attribute__((ext_vector_type(4))) float v4f;
typedef __attribute__((ext_vector_type(8))) float v8f;

#define B_TOT  32768
#define XCOLS  704      // 10*32 + 20*16 + 64
#define XPAD   708      // padded LDS row stride (708 % 64 == 4 -> conflict-free)
#define DM     128
#define NOUT   31
#define SELF0  640      // column offset of self obs inside an x row
#define LSELF  64

__global__ __launch_bounds__(256)
void self_embed_kernel(const float* __restrict__ x,
                       const float* __restrict__ W0, const float* __restrict__ b0,
                       const float* __restrict__ W1, const float* __restrict__ b1,
                       const float* __restrict__ W2, const float* __restrict__ b2,
                       float* __restrict__ out, float* __restrict__ self_out)
{
    __shared__ float lds_x[16 * XPAD];

    const int tid   = threadIdx.x;
    const int bbase = blockIdx.x * 16;

    // ---- stage 16 rows of x into LDS (vec4, coalesced, non-temporal) ------
    {
        const int nvec = 16 * (XCOLS / 4);          // 2816 vec4s
        for (int i = tid; i < nvec; i += 256) {
            const int r  = i / (XCOLS / 4);
            const int c4 = i % (XCOLS / 4);
            v4f v = __builtin_nontemporal_load(
                reinterpret_cast<const v4f*>(x + (size_t)(bbase + r) * XCOLS) + c4);
            reinterpret_cast<v4f*>(&lds_x[r * XPAD])[c4] = v;
        }
    }
    __syncthreads();

    // ---- copy self_x slab to second output (non-temporal stream) ---------
    for (int i = tid; i < 16 * LSELF; i += 256) {
        const int r = i >> 6, c = i & 63;
        __builtin_nontemporal_store(lds_x[r * XPAD + SELF0 + c],
                                    &self_out[(size_t)(bbase + r) * LSELF + c]);
    }

    // ---- per-wave WMMA section -------------------------------------------
    const int wave = tid >> 5;         // 0..7 : 16-wide N slice of d_model
    const int lane = tid & 31;
    const int m    = lane & 15;        // M row within tile (A) / N col (B,C)
    const int kh   = lane >> 4;        // half-wave selects K pair / M+8
    const int koff = kh * 2;           // K offset for A/B fragments
    const int jn   = wave * 16 + m;    // global output column

    // A fragment: rows = 16 staged batch rows, K cols [col, col+3]
    auto ldA = [&](int col) -> v2f {
        return *reinterpret_cast<const v2f*>(&lds_x[m * XPAD + col + koff]);
    };
    // B fragment: W rows [kbase, kbase+3], output col jn (RT: stays cached)
    auto ldB = [&](const float* __restrict__ W, int kbase) -> v2f {
        v2f b;
        b.x = W[(kbase + koff + 0) * DM + jn];
        b.y = W[(kbase + koff + 1) * DM + jn];
        return b;
    };
    auto mma = [&](v2f a, v2f b, v8f c) -> v8f {
        return __builtin_amdgcn_wmma_f32_16x16x4_f32(
            false, a, false, b, (short)0, c, false, false);
    };
    auto splat = [&](float bv) -> v8f {
        return v8f{bv, bv, bv, bv, bv, bv, bv, bv};
    };
    // C/D tile: VGPR v -> M = v + kh*8, N = jn ; write-once stream -> NT
    auto store_tile = [&](int jrow, v8f acc) {
        #pragma unroll
        for (int v = 0; v < 8; ++v) {
            const size_t brow = (size_t)(bbase + v + kh * 8);
            __builtin_nontemporal_store(acc[v],
                &out[(brow * NOUT + jrow) * DM + jn]);
        }
    };

    // ================= group 0 : K_xi=32, W0 is [96,128] ===================
    {
        v8f accS = splat(b0[jn]);
        #pragma unroll
        for (int kk = 0; kk < 16; ++kk)          // self part, K=64
            accS = mma(ldA(SELF0 + kk * 4), ldB(W0, 32 + kk * 4), accS);

        v2f w0a[8];
        #pragma unroll
        for (int kk = 0; kk < 8; ++kk) w0a[kk] = ldB(W0, kk * 4);

        for (int k = 0; k < 10; ++k) {
            v8f acc = accS;
            #pragma unroll
            for (int kk = 0; kk < 8; ++kk)       // xi part, K=32
                acc = mma(ldA(k * 32 + kk * 4), w0a[kk], acc);
            store_tile(k, acc);
        }
    }

    // ================= group 1 : K_xi=16, W1 is [80,128] ===================
    {
        v8f accS = splat(b1[jn]);
        #pragma unroll
        for (int kk = 0; kk < 16; ++kk)          // self part, K=64
            accS = mma(ldA(SELF0 + kk * 4), ldB(W1, 16 + kk * 4), accS);

        v2f w1a[4];
        #pragma unroll
        for (int kk = 0; kk < 4; ++kk) w1a[kk] = ldB(W1, kk * 4);

        for (int k = 0; k < 20; ++k) {
            v8f acc = accS;
            #pragma unroll
            for (int kk = 0; kk < 4; ++kk)       // xi part, K=16
                acc = mma(ldA(320 + k * 16 + kk * 4), w1a[kk], acc);
            store_tile(10 + k, acc);
        }
    }

    // ================= group 2 : self only, W2 is [64,128] =================
    {
        v8f accS = splat(b2[jn]);
        #pragma unroll
        for (int kk = 0; kk < 16; ++kk)
            accS = mma(ldA(SELF0 + kk * 4), ldB(W2, kk * 4), accS);
        store_tile(30, accS);
    }
}

extern "C" void kernel_launch(void* const* d_in, const int* in_sizes, int n_in,
                              void* d_out, int out_size, void* d_ws, size_t ws_size,
                              hipStream_t stream)
{
    const float* x  = (const float*)d_in[0];
    const float* W0 = (const float*)d_in[1];
    const float* b0 = (const float*)d_in[2];
    const float* W1 = (const float*)d_in[3];
    const float* b1 = (const float*)d_in[4];
    const float* W2 = (const float*)d_in[5];
    const float* b2 = (const float*)d_in[6];
    // d_in[7] = self_idx (constant 2, baked into the kernel)

    float* out      = (float*)d_out;
    float* self_out = out + (size_t)B_TOT * NOUT * DM;

    self_embed_kernel<<<B_TOT / 16, 256, 0, stream>>>(
        x, W0, b0, W1, b1, W2, b2, out, self_out);
}